// GraphDiffusionConv_63204738727940
// MI455X (gfx1250) — compile-verified
//
#include <hip/hip_runtime.h>

typedef float v2f __attribute__((ext_vector_type(2)));
typedef float v8f __attribute__((ext_vector_type(8)));

#define NN 8192
#define NQ 16            // B*2 columns of the state matrix
#define KTOT 5
#define SEGS 8
#define KSEG (NN / SEGS)        // 1024
#define CHUNK 64
#define NCHUNK (KSEG / CHUNK)   // 16
#define ROWS_PER_BLOCK 128
#define APAD 68                 // padded LDS row stride (words) for A tile

// ---------------- pack x0: X0[j*16 + b*2 + c] ----------------
__global__ void pack_x0_kernel(const float* __restrict__ in,
                               const float* __restrict__ hid,
                               float* __restrict__ X0) {
    int t = blockIdx.x * blockDim.x + threadIdx.x;   // < N*16
    int j = t >> 4;
    int q = t & 15;
    int b = q >> 1;
    int c = q & 1;
    X0[t] = c ? hid[b * NN + j] : in[b * NN + j];
}

// CDNA5 async global->LDS copies (ASYNCcnt-tracked, no VGPR round-trip).
// Low 32 bits of a flat pointer into __shared__ are the LDS byte offset.
__device__ __forceinline__ void async_g2l_b128(const float* g, const float* l) {
    unsigned loff = (unsigned)(uintptr_t)l;
    asm volatile("global_load_async_to_lds_b128 %0, %1, off"
                 :: "v"(loff), "v"(g) : "memory");
}
__device__ __forceinline__ void async_g2l_b32(const float* g, const float* l) {
    unsigned loff = (unsigned)(uintptr_t)l;
    asm volatile("global_load_async_to_lds_b32 %0, %1, off"
                 :: "v"(loff), "v"(g) : "memory");
}

// ---------------- K-split GEMM: P[seg] = K[rows, kseg] * X[kseg, 16] ----------------
__global__ __launch_bounds__(256) void gemm_seg_kernel(const float* __restrict__ Kmat,
                                                       const float* __restrict__ X,
                                                       float* __restrict__ P) {
    __shared__ float klds[2][ROWS_PER_BLOCK * APAD];  // double-buffered 128x64 A tile
    __shared__ float xlds[2][(CHUNK / 2) * 32];       // double-buffered 64x16 X chunk

    const int t    = threadIdx.x;
    const int lane = t & 31;
    const int wave = t >> 5;           // 0..7
    const int h    = lane >> 4;        // half: 0 or 1
    const int mn   = lane & 15;        // row (A) / col (B,D) index
    const int rowBase = blockIdx.x * ROWS_PER_BLOCK;
    const int segBase = blockIdx.y * KSEG;

    // Issue one chunk's async copies: 8x b128 (A) + 4x b32 (X) per thread.
    // All 256 threads participate in both => ASYNCcnt advances uniformly (12/chunk).
    auto issue_chunk = [&](int ch, int buf) {
        const int kb = segBase + ch * CHUNK;
        #pragma unroll
        for (int i = 0; i < 8; ++i) {
            int f  = i * 256 + t;
            int r  = f >> 4;         // A tile row 0..127
            int c4 = f & 15;         // float4 within row
            const float* g = Kmat + (size_t)(rowBase + r) * NN + kb + c4 * 4;
            async_g2l_b128(g, &klds[buf][r * APAD + c4 * 4]);
        }
        #pragma unroll
        for (int i = 0; i < 4; ++i) {
            int w = i * 256 + t;     // flat word 0..1023
            int j = w >> 4;          // X row 0..63
            int n = w & 15;          // X col
            const float* g = X + (size_t)(kb + j) * NQ + n;
            // row-pair interleave: (j,n) -> word (j>>1)*32 + n*2 + (j&1)
            async_g2l_b32(g, &xlds[buf][(j >> 1) * 32 + n * 2 + (j & 1)]);
        }
    };

    v8f acc = {};

    issue_chunk(0, 0);
    for (int ch = 0; ch < NCHUNK; ++ch) {
        const int cur = ch & 1;
        if (ch + 1 < NCHUNK) {
            issue_chunk(ch + 1, cur ^ 1);                       // prefetch next chunk
            asm volatile("s_wait_asynccnt 0xc" ::: "memory");   // oldest 12 (chunk ch) done
        } else {
            asm volatile("s_wait_asynccnt 0x0" ::: "memory");
        }
        __syncthreads();

        // A frag: lane(h,mn) holds A[mn][kk+2h], A[mn][kk+2h+1]  (16x4 f32 layout)
        // B frag: lane(h,mn) holds B[kk+2h][mn], B[kk+2h+1][mn]
        const float* arow = &klds[cur][(wave * 16 + mn) * APAD + 2 * h];
        const float* brow = &xlds[cur][h * 32 + mn * 2];
        #pragma unroll
        for (int kk = 0; kk < CHUNK; kk += 4) {
            v2f a = *(const v2f*)(arow + kk);
            v2f b = *(const v2f*)(brow + (kk >> 1) * 32);
            acc = __builtin_amdgcn_wmma_f32_16x16x4_f32(
                /*neg_a=*/false, a, /*neg_b=*/false, b,
                /*c_mod=*/(short)0, acc, /*reuse_a=*/false, /*reuse_b=*/false);
        }
        __syncthreads();   // all waves done reading buf `cur` before it is re-targeted
    }

    // D layout: VGPR r holds row (r + 8h), col = mn
    float* dst = P + (size_t)blockIdx.y * (NN * NQ)
                   + (size_t)(rowBase + wave * 16 + 8 * h) * NQ + mn;
    #pragma unroll
    for (int r = 0; r < 8; ++r)
        dst[r * NQ] = acc[r];
}

// ---------------- deterministic segment reduce + alpha/beta ----------------
__global__ void reduce_axpby_kernel(const float* __restrict__ P,
                                    const float* __restrict__ prev,
                                    float* __restrict__ out,
                                    float alpha, float beta) {
    int t = blockIdx.x * blockDim.x + threadIdx.x;   // < N*16
    float s = 0.0f;
    #pragma unroll
    for (int sg = 0; sg < SEGS; ++sg)
        s += P[(size_t)sg * (NN * NQ) + t];
    out[t] = alpha * s + beta * prev[t];
}

// ---------------- epilogue (bug-compatible reshape + Linear) ----------------
__global__ void epilogue_kernel(const float* __restrict__ X0,
                                const float* __restrict__ X10,
                                const float* __restrict__ X20,
                                const float* __restrict__ X11,
                                const float* __restrict__ X21,
                                const float* __restrict__ W,
                                const float* __restrict__ bias,
                                float* __restrict__ out) {
    int t = blockIdx.x * blockDim.x + threadIdx.x;   // < B*N = 65536
    int b2 = t >> 13;          // output batch
    int i  = t & 8191;         // output node
    int b  = i >> 10;          // source batch of the raw-reinterpreted element
    int c  = b2 & 1;
    int q  = b * 2 + c;
    const float* Xs[KTOT] = {X0, X10, X20, X11, X21};
    float s = bias[0];
    #pragma unroll
    for (int c2 = 0; c2 < 2; ++c2) {
        int j = ((i & 1023) << 3) + (c2 << 2) + (b2 >> 1);
        #pragma unroll
        for (int k = 0; k < KTOT; ++k)
            s += Xs[k][j * NQ + q] * W[c2 * KTOT + k];
    }
    out[t] = s;
}

extern "C" void kernel_launch(void* const* d_in, const int* in_sizes, int n_in,
                              void* d_out, int out_size, void* d_ws, size_t ws_size,
                              hipStream_t stream) {
    const float* input  = (const float*)d_in[0];
    const float* hidden = (const float*)d_in[1];
    const float* ker0   = (const float*)d_in[2];
    const float* ker1   = (const float*)d_in[3];
    const float* W      = (const float*)d_in[4];
    const float* bptr   = (const float*)d_in[5];
    float* out = (float*)d_out;
    float* ws  = (float*)d_ws;

    const size_t XSZ = (size_t)NN * NQ;   // 131072 floats
    float* X0 = ws;
    float* Xk[4] = { ws + XSZ, ws + 2 * XSZ, ws + 3 * XSZ, ws + 4 * XSZ };
    float* P  = ws + 5 * XSZ;             // SEGS * XSZ floats of partials

    pack_x0_kernel<<<(NN * NQ) / 256, 256, 0, stream>>>(input, hidden, X0);

    const float* kers[2] = { ker0, ker1 };
    for (int kn = 0; kn < 2; ++kn) {
        float* X1 = Xk[kn * 2];
        float* X2 = Xk[kn * 2 + 1];
        // x1 = K @ x0
        gemm_seg_kernel<<<dim3(NN / ROWS_PER_BLOCK, SEGS), 256, 0, stream>>>(kers[kn], X0, P);
        reduce_axpby_kernel<<<(NN * NQ) / 256, 256, 0, stream>>>(P, X0, X1, 1.0f, 0.0f);
        // x2 = 2*K @ x1 - x0
        gemm_seg_kernel<<<dim3(NN / ROWS_PER_BLOCK, SEGS), 256, 0, stream>>>(kers[kn], X1, P);
        reduce_axpby_kernel<<<(NN * NQ) / 256, 256, 0, stream>>>(P, X0, X2, 2.0f, -1.0f);
    }

    epilogue_kernel<<<(NN * 8) / 256, 256, 0, stream>>>(
        X0, Xk[0], Xk[1], Xk[2], Xk[3], W, bptr, out);
}